// DdiModel_61830349193388
// MI455X (gfx1250) — compile-verified
//
#include <hip/hip_runtime.h>

#define D 128

typedef __attribute__((ext_vector_type(2))) float v2f;
typedef __attribute__((ext_vector_type(8))) float v8f;

// ---------------------------------------------------------------- utilities

__global__ void fill_kernel(float* p, float v, long n) {
  long i = (long)blockIdx.x * blockDim.x + threadIdx.x;
  if (i < n) p[i] = v;
}

// Pack row-major W[K][Ncols] into K-pair-interleaved Wp[(K/2)][Ncols][2]:
// Wp[kp][n] = { W[2kp][n], W[2kp+1][n] }  -> one b64 load per WMMA B-fragment.
__global__ void pack_pairs_kernel(const float* __restrict__ W, float* __restrict__ Wp,
                                  int K, int Ncols) {
  int i = blockIdx.x * blockDim.x + threadIdx.x;
  int total = (K >> 1) * Ncols;
  if (i >= total) return;
  int kp = i / Ncols, n = i - kp * Ncols;
  float2 v;
  v.x = W[(long)(2 * kp) * Ncols + n];
  v.y = W[(long)(2 * kp + 1) * Ncols + n];
  ((float2*)Wp)[(long)kp * Ncols + n] = v;
}

// deg_s[n] = 1 + #times n appears as sender; deg_r likewise for receivers.
__global__ void degree_kernel(const int* __restrict__ snd, const int* __restrict__ rcv,
                              float* deg_s, float* deg_r, int E) {
  int e = blockIdx.x * blockDim.x + threadIdx.x;
  if (e < E) {
    unsafeAtomicAdd(&deg_s[snd[e]], 1.0f);
    unsafeAtomicAdd(&deg_r[rcv[e]], 1.0f);
  }
}

// dst[row][:] = src[row][:] * rsqrt(deg[row]); float4 vectorized over N*D.
__global__ void rownorm_kernel(const float* __restrict__ src, const float* __restrict__ deg,
                               float* __restrict__ dst, int n_nodes) {
  long i = (long)blockIdx.x * blockDim.x + threadIdx.x; // float4 index
  long n4 = (long)n_nodes * (D / 4);
  if (i >= n4) return;
  int row = (int)(i >> 5);                               // (i*4)/128
  float s = rsqrtf(deg[row]);
  float4 v = ((const float4*)src)[i];
  v.x *= s; v.y *= s; v.z *= s; v.w *= s;
  ((float4*)dst)[i] = v;
}

// msum[rcv[e]][:] += xn[snd[e]][:]  — one thread per (edge, 4-float chunk).
__global__ void scatter_kernel(const int* __restrict__ snd, const int* __restrict__ rcv,
                               const float* __restrict__ xn, float* msum, int E) {
  long gid = (long)blockIdx.x * blockDim.x + threadIdx.x;
  long total = (long)E * 32;
  if (gid >= total) return;
  int e = (int)(gid >> 5);
  int c = (int)(gid & 31) << 2;
  int s = snd[e], r = rcv[e];
  float4 v = *(const float4*)(xn + (long)s * D + c);
  float* dst = msum + (long)r * D + c;
  unsafeAtomicAdd(dst + 0, v.x);
  unsafeAtomicAdd(dst + 1, v.y);
  unsafeAtomicAdd(dst + 2, v.z);
  unsafeAtomicAdd(dst + 3, v.w);
}

// ------------------------------------------------------- fused SAGE GEMM
// out[m][:] = act( concat(A0[m], (msum[m]+xn[m])*deg_r[m]^-1.5) @ W + b )
// One wave per 16-row M-tile; wave holds all 8 N-tiles (64 f32 acc VGPRs),
// so out may alias msum (each wave reads/writes only its own rows).
// K-loop is split into two branch-free halves (self | aggregated).
// B fragments come from the pair-packed Wp (single global_load_b64 each).

__global__ __launch_bounds__(256)
void sage_gemm_kernel(const float* __restrict__ A0,   // [N,128] self features
                      const float* msum,              // [N,128] scatter sums (may alias out)
                      const float* xn,                // [N,128] normalized self (self-loop msg)
                      const float* __restrict__ deg_r,// [N]
                      const float* __restrict__ Wp,   // [128][128][2] pair-packed W
                      const float* __restrict__ bias, // [128]
                      float* out,                     // [N,128] (may alias msum)
                      int n_nodes, int do_relu) {
  int wave = (blockIdx.x << 3) + (threadIdx.x >> 5);
  int mt_count = (n_nodes + 15) >> 4;
  if (wave >= mt_count) return;
  int lane = threadIdx.x & 31;
  int lo = lane & 15, hi = lane >> 4;

  int m = (wave << 4) + lo;          // A-matrix row held by this lane (both halves)
  bool mvalid = m < n_nodes;
  int mr = mvalid ? m : 0;

  float c = deg_r[mr];
  float rs = rsqrtf(c);
  float scale = rs * rs * rs;        // (1/c) * rsqrt(c) = c^-1.5

  const float* a0row = A0 + (long)mr * D;
  const float* smrow = msum + (long)mr * D;
  const float* xnrow = xn + (long)mr * D;
  const v2f* wp = (const v2f*)Wp;    // [kp][n] pairs

  v8f acc[8];
#pragma unroll
  for (int t = 0; t < 8; ++t) acc[t] = (v8f)0.0f;

  // ---- K in [0,128): A from self features (branch-free)
  for (int kk = 0; kk < D; kk += 4) {
    int k0 = kk + (hi << 1);         // lanes 16-31 carry K+2,K+3
    v2f a = *(const v2f*)(a0row + k0);
    if (!mvalid) { a.x = 0.0f; a.y = 0.0f; }
    const v2f* wrow = wp + (long)(k0 >> 1) * D + lo;
#pragma unroll
    for (int t = 0; t < 8; ++t) {
      v2f b = wrow[t * 16];
      acc[t] = __builtin_amdgcn_wmma_f32_16x16x4_f32(
          false, a, false, b, (short)0, acc[t], false, false);
    }
  }

  // ---- K in [128,256): A = (msum + self_msg) * deg_r^-1.5 (branch-free)
  for (int kk = 0; kk < D; kk += 4) {
    int k0 = kk + (hi << 1);
    v2f sm = *(const v2f*)(smrow + k0);
    v2f xv = *(const v2f*)(xnrow + k0);
    v2f a;
    a.x = (sm.x + xv.x) * scale;
    a.y = (sm.y + xv.y) * scale;
    if (!mvalid) { a.x = 0.0f; a.y = 0.0f; }
    const v2f* wrow = wp + (long)((k0 + D) >> 1) * D + lo;
#pragma unroll
    for (int t = 0; t < 8; ++t) {
      v2f b = wrow[t * 16];
      acc[t] = __builtin_amdgcn_wmma_f32_16x16x4_f32(
          false, a, false, b, (short)0, acc[t], false, false);
    }
  }

#pragma unroll
  for (int t = 0; t < 8; ++t) {
    int col = t * 16 + lo;
    float bv = bias[col];
#pragma unroll
    for (int v = 0; v < 8; ++v) {
      int row = (wave << 4) + v + (hi << 3);   // C/D: M = v + 8*hi
      float val = acc[t][v] + bv;
      if (do_relu) val = fmaxf(val, 0.0f);
      if (row < n_nodes) out[(long)row * D + col] = val;
    }
  }
}

// ------------------------------------------------------- pair MLP
// score[p] = relu( (h[i]*h[j]) @ Wa + ba ) @ Wb + bb
// One wave per 16-pair tile. z tile staged in LDS (stride 132 floats:
// 16B-aligned float4 rows, conflict-free b64 WMMA A-frag reads).

#define ZSTRIDE 132

__global__ __launch_bounds__(128)
void pair_mlp_kernel(const int* __restrict__ pairs, const float* __restrict__ h,
                     const float* __restrict__ Wpa,  // [64][128][2] pair-packed Wa
                     const float* __restrict__ ba,
                     const float* __restrict__ Wb, const float* __restrict__ bb,
                     float* __restrict__ out, int n_pairs) {
  __shared__ float zsh[4][16 * ZSTRIDE];
  int wv = threadIdx.x >> 5;
  int tile = (blockIdx.x << 2) + wv;
  int ntiles = (n_pairs + 15) >> 4;
  if (tile >= ntiles) return;
  int lane = threadIdx.x & 31;
  int lo = lane & 15, hi = lane >> 4;
  float* z = zsh[wv];

  // stage z = h[i]*h[j]: lane handles pair `lo`, half `hi` (64 floats)
  {
    int p = (tile << 4) + lo;
    int pc = p < n_pairs ? p : 0;
    int i = pairs[2 * pc], j = pairs[2 * pc + 1];
    const float4* hi4 = (const float4*)(h + (long)i * D);
    const float4* hj4 = (const float4*)(h + (long)j * D);
    float4* zrow = (float4*)(z + lo * ZSTRIDE);   // 16B aligned: 132*4*lo % 16 == 0
    for (int q = hi * 16; q < hi * 16 + 16; ++q) {
      float4 a = hi4[q], b = hj4[q];
      float4 v;
      v.x = a.x * b.x; v.y = a.y * b.y; v.z = a.z * b.z; v.w = a.w * b.w;
      zrow[q] = v;
    }
  }
  // wave reads back only its own LDS writes; compiler inserts s_wait_dscnt.

  const v2f* wp = (const v2f*)Wpa;
  float partial[8];
#pragma unroll
  for (int v = 0; v < 8; ++v) partial[v] = 0.0f;

  for (int t = 0; t < 8; ++t) {
    v8f acc = (v8f)0.0f;
    for (int kk = 0; kk < D; kk += 4) {
      int k0 = kk + (hi << 1);
      v2f a = *(const v2f*)(z + lo * ZSTRIDE + k0);   // ds_load_b64, conflict-free
      v2f b = wp[(long)(k0 >> 1) * D + t * 16 + lo];  // global_load_b64
      acc = __builtin_amdgcn_wmma_f32_16x16x4_f32(
          false, a, false, b, (short)0, acc, false, false);
    }
    int col = t * 16 + lo;
    float bv = ba[col], wb = Wb[col];
#pragma unroll
    for (int v = 0; v < 8; ++v) {
      partial[v] += fmaxf(acc[v] + bv, 0.0f) * wb;
    }
  }

  // reduce over the 16 lanes of each half (cols 0..15)
#pragma unroll
  for (int v = 0; v < 8; ++v) {
    float x = partial[v];
    x += __shfl_xor(x, 1, 32);
    x += __shfl_xor(x, 2, 32);
    x += __shfl_xor(x, 4, 32);
    x += __shfl_xor(x, 8, 32);
    partial[v] = x;
  }
  if (lo == 0) {
    float bbv = bb[0];
#pragma unroll
    for (int v = 0; v < 8; ++v) {
      int row = (tile << 4) + v + (hi << 3);   // M = v + 8*hi
      if (row < n_pairs) out[row] = partial[v] + bbv;
    }
  }
}

// ---------------------------------------------------------------- launcher

extern "C" void kernel_launch(void* const* d_in, const int* in_sizes, int n_in,
                              void* d_out, int out_size, void* d_ws, size_t ws_size,
                              hipStream_t stream) {
  // inputs: 0 node_ids (identity, unused), 1 senders, 2 receivers, 3 pairs,
  //         4 emb, 5 W1, 6 b1, 7 W2, 8 b2, 9 Wa, 10 ba, 11 Wb, 12 bb
  const int* senders   = (const int*)d_in[1];
  const int* receivers = (const int*)d_in[2];
  const int* pairs     = (const int*)d_in[3];
  const float* emb = (const float*)d_in[4];
  const float* W1  = (const float*)d_in[5];
  const float* b1  = (const float*)d_in[6];
  const float* W2  = (const float*)d_in[7];
  const float* b2  = (const float*)d_in[8];
  const float* Wa  = (const float*)d_in[9];
  const float* ba  = (const float*)d_in[10];
  const float* Wb  = (const float*)d_in[11];
  const float* bb  = (const float*)d_in[12];
  float* out = (float*)d_out;

  int N = in_sizes[0];
  int E = in_sizes[1];
  int P = in_sizes[3] / 2;
  long ND = (long)N * D;

  float* ws    = (float*)d_ws;
  float* deg_s = ws;
  float* deg_r = ws + N;
  float* bufA  = ws + 2L * N;    // xn / xn2
  float* bufB  = bufA + ND;      // msum1 -> x1   (GEMM out aliases msum: row-owned, safe)
  float* bufC  = bufB + ND;      // msum2 -> h
  float* Wp1   = bufC + ND;      // 256x128 pair-packed  (32768 floats)
  float* Wp2   = Wp1 + 2L * D * D;
  float* Wpa   = Wp2 + 2L * D * D; // 128x128 pair-packed (16384 floats)

  const int T = 256;
  // init: degrees start at 1.0 (self-loop), message sums at 0
  fill_kernel<<<(N + T - 1) / T, T, 0, stream>>>(deg_s, 1.0f, N);
  fill_kernel<<<(N + T - 1) / T, T, 0, stream>>>(deg_r, 1.0f, N);
  fill_kernel<<<(int)((ND + T - 1) / T), T, 0, stream>>>(bufB, 0.0f, ND);
  fill_kernel<<<(int)((ND + T - 1) / T), T, 0, stream>>>(bufC, 0.0f, ND);

  // pre-pack weights into K-pair-interleaved layout (B fragment = 1x b64)
  pack_pairs_kernel<<<(2 * D * D / 2 + T - 1) / T, T, 0, stream>>>(W1, Wp1, 2 * D, D);
  pack_pairs_kernel<<<(2 * D * D / 2 + T - 1) / T, T, 0, stream>>>(W2, Wp2, 2 * D, D);
  pack_pairs_kernel<<<(D * D / 2 + T - 1) / T, T, 0, stream>>>(Wa, Wpa, D, D);

  degree_kernel<<<(E + T - 1) / T, T, 0, stream>>>(senders, receivers, deg_s, deg_r, E);

  long n4 = ND / 4;
  long sth = (long)E * 32;
  int mt = (N + 15) >> 4;

  // ---- layer 1: x1 = relu(sage(emb))
  rownorm_kernel<<<(int)((n4 + T - 1) / T), T, 0, stream>>>(emb, deg_s, bufA, N);
  scatter_kernel<<<(int)((sth + T - 1) / T), T, 0, stream>>>(senders, receivers, bufA, bufB, E);
  sage_gemm_kernel<<<(mt + 7) / 8, 256, 0, stream>>>(emb, bufB, bufA, deg_r, Wp1, b1,
                                                     bufB, N, /*relu=*/1);

  // ---- layer 2: h = sage(x1)
  rownorm_kernel<<<(int)((n4 + T - 1) / T), T, 0, stream>>>(bufB, deg_s, bufA, N);
  scatter_kernel<<<(int)((sth + T - 1) / T), T, 0, stream>>>(senders, receivers, bufA, bufC, E);
  sage_gemm_kernel<<<(mt + 7) / 8, 256, 0, stream>>>(bufB, bufC, bufA, deg_r, Wp2, b2,
                                                     bufC, N, /*relu=*/0);

  // ---- link predictor
  int ptiles = (P + 15) >> 4;
  pair_mlp_kernel<<<(ptiles + 3) / 4, 128, 0, stream>>>(pairs, bufC, Wpa, ba, Wb, bb, out, P);
}